// ShiftedProxActivation_39273180955313
// MI455X (gfx1250) — compile-verified
//
#include <hip/hip_runtime.h>
#include <hip/hip_bf16.h>
#include <math.h>

// ---------------------------------------------------------------------------
// Primal-dual graph solver for MI455X (gfx1250).
// Bandwidth/latency-bound (no matmul structure): uses CDNA5 async
// global->LDS staging (ASYNCcnt), global_prefetch, wave32 shuffles,
// and f32 global atomics. Working set (~130MB) is L2-resident (192MB).
// ---------------------------------------------------------------------------

#define EPB        256   // edges per block
#define DFEAT      32    // feature dim (fixed by reference)
#define NBASIS     8
#define NNEWTON    8
#define EPSR       1e-8f
#define NUM_PD     12

__device__ __forceinline__ float softplus_f(float x) {
  return log1pf(__expf(x));
}

// ---- tiny setup: constrained params + scalars derived from max_degree -----
__global__ void setup_kernel(const float* __restrict__ raw_a,
                             const float* __restrict__ raw_beta,
                             const float* __restrict__ raw_b0,
                             const float* __restrict__ c,
                             const int*   __restrict__ md_ptr,
                             float* __restrict__ params) {
  if (threadIdx.x != 0 || blockIdx.x != 0) return;
  float md  = (float)md_ptr[0];
  float tau = 0.9f / sqrtf(2.0f * md);          // KAPPA / sqrt(2*max_degree)
  params[0] = tau;                               // sigma == tau
  params[1] = 1.0f / tau;                        // inv_sigma
  params[2] = 1.0f / (1.0f + tau);               // inv_denom (z_coeff = tau)
  float b0   = softplus_f(raw_b0[0]);
  float psi0 = b0;
  for (int k = 0; k < NBASIS; ++k) {
    float a  = softplus_f(raw_a[k]);
    float be = softplus_f(raw_beta[k]) + 1e-4f;
    float cc = c[k];
    psi0 += a / (1.0f + __expf(-cc));            // a * sigmoid(c)
    params[8  + k] = a;
    params[16 + k] = a * be;
    params[24 + k] = be;
    params[32 + k] = cc;
  }
  params[3] = psi0;
  params[4] = b0;
}

// ---- edge kernel: y_tilde -> masked Newton prox -> y_next + scatter -------
__global__ __launch_bounds__(256) void edge_kernel(
    const float* __restrict__ ubar,
    float*       __restrict__ y,          // in/out (in place)
    const float* __restrict__ offsets,
    const float* __restrict__ weights,
    const int*   __restrict__ rowp,
    const int*   __restrict__ colp,
    const float* __restrict__ params,
    float*       __restrict__ adj,
    int E)
{
  __shared__ __align__(16) float s_tile[EPB * DFEAT]; // offsets tile -> y_tilde (in place)
  __shared__ float s_r[EPB];                          // r^2 -> (1 - scale)
  __shared__ float s_p[64];                           // params

  const int tid  = threadIdx.x;
  const int lane = tid & 31;
  const int wv   = tid >> 5;
  const int base = blockIdx.x * EPB;

  if (tid < 64) s_p[tid] = params[tid];

  // --- async-stage the streaming offsets tile (EPB*DFEAT*4 = 32KB) into LDS.
  // 8 rounds of b128 per thread; addresses clamped in-bounds for the tail block.
  {
    unsigned  lds0 = (unsigned)(size_t)(&s_tile[0]) + (unsigned)(tid * 16);
    long long gmax = (long long)E * (DFEAT * 4) - 16;
    #pragma unroll
    for (int k = 0; k < 8; ++k) {
      long long gb = (long long)base * (DFEAT * 4) + (long long)(tid * 16 + k * 4096);
      if (gb > gmax) gb = gmax;
      unsigned long long ga = (unsigned long long)(size_t)((const char*)offsets + gb);
      unsigned lds = lds0 + (unsigned)(k * 4096);
      asm volatile("global_load_async_to_lds_b128 %0, %1, off"
                   :: "v"(lds), "v"(ga) : "memory");
    }
  }

  // per-lane edge metadata for this wave's 32 edges (edge = base + wv*32 + lane)
  int   myE   = base + wv * 32 + lane;
  int   myEc  = (myE < E) ? myE : (E - 1);
  int   row_v = rowp[myEc];
  int   col_v = colp[myEc];
  float w_v   = weights[myEc];

  asm volatile("s_wait_asynccnt 0x0" ::: "memory");
  __syncthreads();

  const float sigma = s_p[0];

  // --- phase 1: y_tilde into LDS (in place over offsets), ||y_tilde||^2 per edge
  for (int i = 0; i < 32; ++i) {
    int eL = wv * 32 + i;
    int e  = base + eL;
    float yt = 0.0f;
    if (e < E) {
      if (i + 2 < 32) { // prefetch gather targets two edges ahead (L2-hot u_bar)
        int rn = __shfl(row_v, i + 2);
        int cn = __shfl(col_v, i + 2);
        __builtin_prefetch(&ubar[rn * DFEAT + lane], 0, 3);
        __builtin_prefetch(&ubar[cn * DFEAT + lane], 0, 3);
      }
      int r_ = __shfl(row_v, i);
      int c_ = __shfl(col_v, i);
      float g   = ubar[r_ * DFEAT + lane] - ubar[c_ * DFEAT + lane];
      float yv  = y[e * DFEAT + lane];
      float off = s_tile[eL * DFEAT + lane];
      yt = yv + sigma * (g - off);
    }
    s_tile[eL * DFEAT + lane] = yt;
    float r2 = yt * yt;
    #pragma unroll
    for (int dlt = 16; dlt > 0; dlt >>= 1) r2 += __shfl_xor(r2, dlt);
    if (lane == 0) s_r[eL] = r2;
  }
  __syncthreads();

  // --- phase 2: masked Newton, one thread per edge (no lane redundancy)
  {
    const float inv_sigma = s_p[1];
    const float psi0      = s_p[3];
    const float b0        = s_p[4];
    float A[8], AB[8], Bt[8], Ct[8];
    #pragma unroll
    for (int k = 0; k < 8; ++k) {
      A[k]  = s_p[8 + k];
      AB[k] = s_p[16 + k];
      Bt[k] = s_p[24 + k];
      Ct[k] = s_p[32 + k];
    }
    float r   = sqrtf(s_r[tid]) * inv_sigma;   // ||y_tilde/sigma||
    float lam = w_v * inv_sigma;
    bool  active = (myE < E) && (r > EPSR) && (lam * psi0 - r < 0.0f);
    float s = active ? r : 0.0f;
    #pragma unroll
    for (int n = 0; n < NNEWTON; ++n) {
      float p = b0, pp = 0.0f;
      #pragma unroll
      for (int k = 0; k < 8; ++k) {
        float u  = s * Bt[k] + Ct[k];          // s >= 0 always (clamped)
        float sg = 1.0f / (1.0f + __expf(-u));
        p  += A[k] * sg;
        pp += AB[k] * sg * (1.0f - sg);
      }
      float sn = s - (s + lam * p - r) / (1.0f + lam * pp);
      sn = fminf(fmaxf(sn, 0.0f), r);
      s  = active ? sn : s;
    }
    float scale = (r > EPSR) ? (s / r) : 0.0f;
    s_r[tid] = 1.0f - scale;                   // y_next = y_tilde * (1 - scale)
  }
  __syncthreads();

  // --- phase 3: rescale, write y, scatter-add +/- into adj
  for (int i = 0; i < 32; ++i) {
    int eL = wv * 32 + i;
    int e  = base + eL;
    if (e >= E) break;
    int   r_ = __shfl(row_v, i);
    int   c_ = __shfl(col_v, i);
    float f  = s_r[eL];
    float yn = s_tile[eL * DFEAT + lane] * f;
    y[e * DFEAT + lane] = yn;
    atomicAdd(&adj[r_ * DFEAT + lane],  yn);
    atomicAdd(&adj[c_ * DFEAT + lane], -yn);
  }
}

// ---- node kernel: u/u_bar update (float4), re-zero adj for next iteration -
__global__ __launch_bounds__(256) void node_kernel(
    float4*       __restrict__ u,
    float4*       __restrict__ ubar,
    float4*       __restrict__ adj,
    const float4* __restrict__ z,
    const float*  __restrict__ params,
    int ND4)
{
  int idx = blockIdx.x * 256 + threadIdx.x;
  if (idx >= ND4) return;
  float tau       = params[0];
  float inv_denom = params[2];
  float4 uo = u[idx];
  float4 av = adj[idx];
  float4 zv = z[idx];
  float4 un, ub;
  un.x = (uo.x - tau * av.x + tau * zv.x) * inv_denom;
  un.y = (uo.y - tau * av.y + tau * zv.y) * inv_denom;
  un.z = (uo.z - tau * av.z + tau * zv.z) * inv_denom;
  un.w = (uo.w - tau * av.w + tau * zv.w) * inv_denom;
  ub.x = 2.0f * un.x - uo.x;   // u_next + XI*(u_next - u), XI = 1
  ub.y = 2.0f * un.y - uo.y;
  ub.z = 2.0f * un.z - uo.z;
  ub.w = 2.0f * un.w - uo.w;
  u[idx]    = un;
  ubar[idx] = ub;
  adj[idx]  = make_float4(0.0f, 0.0f, 0.0f, 0.0f);
}

// ---------------------------------------------------------------------------
extern "C" void kernel_launch(void* const* d_in, const int* in_sizes, int n_in,
                              void* d_out, int out_size, void* d_ws, size_t ws_size,
                              hipStream_t stream) {
  const float* z        = (const float*)d_in[0];   // [N, 32]
  const float* weights  = (const float*)d_in[1];   // [E]
  const float* offsets  = (const float*)d_in[2];   // [E, 32]
  const float* raw_a    = (const float*)d_in[3];   // [8]
  const float* raw_beta = (const float*)d_in[4];   // [8]
  const float* raw_b0   = (const float*)d_in[5];   // [1]
  const float* c        = (const float*)d_in[6];   // [8]
  const int*   eidx     = (const int*)d_in[7];     // [2, E]
  const int*   md       = (const int*)d_in[8];     // [1]

  const int ND = in_sizes[0];                      // N * 32
  const int E  = in_sizes[1];
  const int* rowp = eidx;
  const int* colp = eidx + E;

  // workspace layout: params(64f) | u_bar(ND) | adj(ND) | y(E*32)
  float* params = (float*)d_ws;
  float* ubar   = params + 64;
  float* adj    = ubar + ND;
  float* y      = adj + ND;
  float* u      = (float*)d_out;                   // u lives in d_out

  setup_kernel<<<1, 64, 0, stream>>>(raw_a, raw_beta, raw_b0, c, md, params);
  hipMemcpyAsync(u,    z, (size_t)ND * sizeof(float), hipMemcpyDeviceToDevice, stream);
  hipMemcpyAsync(ubar, z, (size_t)ND * sizeof(float), hipMemcpyDeviceToDevice, stream);
  hipMemsetAsync(adj, 0, (size_t)ND * sizeof(float), stream);
  hipMemsetAsync(y,   0, (size_t)E * DFEAT * sizeof(float), stream);

  const int gridE = (E + EPB - 1) / EPB;
  const int ND4   = ND / 4;
  const int gridN = (ND4 + 255) / 256;
  for (int it = 0; it < NUM_PD; ++it) {
    edge_kernel<<<gridE, 256, 0, stream>>>(ubar, y, offsets, weights,
                                           rowp, colp, params, adj, E);
    node_kernel<<<gridN, 256, 0, stream>>>((float4*)u, (float4*)ubar, (float4*)adj,
                                           (const float4*)z, params, ND4);
  }
}